// TorchEAM_1400159338716
// MI455X (gfx1250) — compile-verified
//
#include <hip/hip_runtime.h>

#define N_NODES_C   50000
#define N_EDGES_C   3200000
#define N_INT       2000
#define RAD_FLOATS  (N_INT * 8)                 // radial_coeffs: (2000,4,2) = 16000 f32 = 64 KB
#define DR_F        ((float)(6.0 / 2000.0))     // CUTOFF/(NR-1)
#define DRHO_F      ((float)(100.0 / 2000.0))   // RHO_MAX/(NRHO-1)

typedef int v4i __attribute__((vector_size(16)));
typedef __attribute__((address_space(1))) v4i* gptr_v4i;   // non-const: builtin wants mutable ptr type
typedef __attribute__((address_space(3))) v4i* lptr_v4i;

// ---------------------------------------------------------------------------
// Stage the 64KB radial spline table into LDS.
// CDNA5 path: async global->LDS DMA (ASYNCcnt) ; fallback: plain loads.
// ---------------------------------------------------------------------------
__device__ __forceinline__ void stage_radial_lds(float* s, const float* g) {
#if defined(__gfx1250__) && \
    __has_builtin(__builtin_amdgcn_global_load_async_to_lds_b128) && \
    __has_builtin(__builtin_amdgcn_s_wait_asynccnt)
  gptr_v4i gsrc = (gptr_v4i)(float*)(uintptr_t)g;  // cast away const, then to AS1 v4i*
  lptr_v4i ldst = (lptr_v4i)s;
  const int nvec = RAD_FLOATS / 4;  // 4000 x 16B
  for (int i = (int)threadIdx.x; i < nvec; i += (int)blockDim.x) {
    __builtin_amdgcn_global_load_async_to_lds_b128(gsrc + i, ldst + i,
                                                   /*offset=*/0, /*cpol=*/0);
  }
  __builtin_amdgcn_s_wait_asynccnt(0);
#else
  for (int i = threadIdx.x; i < RAD_FLOATS; i += blockDim.x) s[i] = g[i];
#endif
  __syncthreads();
}

// wave32 + block reduction (valid on thread 0)
__device__ __forceinline__ float block_sum256(float v, float* s_red) {
#pragma unroll
  for (int off = 16; off > 0; off >>= 1) v += __shfl_xor(v, off, 32);
  const int lane = threadIdx.x & 31;
  const int wv   = threadIdx.x >> 5;
  if (lane == 0) s_red[wv] = v;
  __syncthreads();
  float t = 0.0f;
  if (wv == 0) {
    t = (lane < (int)(blockDim.x >> 5)) ? s_red[lane] : 0.0f;
#pragma unroll
    for (int off = 16; off > 0; off >>= 1) t += __shfl_xor(t, off, 32);
  }
  return t;
}

// ---------------------------------------------------------------------------
// Pass 0: zero energy+forces output and density scratch (outputs are poisoned
// between calls by the harness, and atomics accumulate into them).
// ---------------------------------------------------------------------------
__global__ void k_zero(float* __restrict__ out, float* __restrict__ density) {
  const int i = blockIdx.x * blockDim.x + threadIdx.x;
  if (i < 1 + 3 * N_NODES_C) out[i] = 0.0f;
  if (i < N_NODES_C) density[i] = 0.0f;
}

// ---------------------------------------------------------------------------
// Pass 1 (edges): bondlen -> radial spline ; density[dst] += rho (L2 atomic);
// energy += sum(rphi / (2*bl))  (one atomic per block).
// ---------------------------------------------------------------------------
__global__ void __launch_bounds__(256)
k_edge_density(const float* __restrict__ r, const float* __restrict__ radial,
               const int* __restrict__ dst, float* __restrict__ density,
               float* __restrict__ energy) {
  __shared__ __align__(16) float s_coef[RAD_FLOATS];
  __shared__ float s_red[8];
  stage_radial_lds(s_coef, radial);

  const int stride = gridDim.x * blockDim.x;
  float phi_acc = 0.0f;
  for (int e = blockIdx.x * blockDim.x + threadIdx.x; e < N_EDGES_C; e += stride) {
    if (e + stride < N_EDGES_C) __builtin_prefetch(r + 3 * (e + stride), 0, 0);
    const float x = r[3 * e + 0], y = r[3 * e + 1], z = r[3 * e + 2];
    const float bl = sqrtf(x * x + y * y + z * z);
    int idx = (int)(bl / DR_F);
    idx = idx < 0 ? 0 : (idx > N_INT - 1 ? N_INT - 1 : idx);
    const float dxv = bl - (float)idx * DR_F;
    const float4* c4 = (const float4*)(s_coef + idx * 8);
    const float4 A = c4[0];  // a_rho a_phi b_rho b_phi
    const float4 B = c4[1];  // c_rho c_phi d_rho d_phi
    const float rho  = A.x + dxv * (A.z + dxv * (B.x + dxv * B.z));
    const float rphi = A.y + dxv * (A.w + dxv * (B.y + dxv * B.w));
    atomicAdd(density + dst[e], rho);
    phi_acc += rphi / (2.0f * bl);
  }
  const float t = block_sum256(phi_acc, s_red);
  if (threadIdx.x == 0) atomicAdd(energy, t);
}

// ---------------------------------------------------------------------------
// Pass 2 (nodes): embedding spline F(density); energy += sum F; store F'.
// ---------------------------------------------------------------------------
__global__ void __launch_bounds__(256)
k_embed(const float* __restrict__ density, const float4* __restrict__ embed4,
        float* __restrict__ Fprime, float* __restrict__ energy) {
  __shared__ float s_red[8];
  const int i = blockIdx.x * blockDim.x + threadIdx.x;
  float F = 0.0f;
  if (i < N_NODES_C) {
    const float d = density[i];
    int idx = (int)(d / DRHO_F);
    idx = idx < 0 ? 0 : (idx > N_INT - 1 ? N_INT - 1 : idx);
    const float dxv = d - (float)idx * DRHO_F;
    const float4 c = embed4[idx];  // a b c d
    F = c.x + dxv * (c.y + dxv * (c.z + dxv * c.w));
    Fprime[i] = c.y + dxv * (2.0f * c.z + 3.0f * c.w * dxv);
  }
  const float t = block_sum256(F, s_red);
  if (threadIdx.x == 0) atomicAdd(energy, t);
}

// ---------------------------------------------------------------------------
// Pass 3 (edges): dE/dr_e = (F'(dst)*rho'(bl) + d/dbl[rphi/(2bl)]) * r/bl ;
// forces[src] += g, forces[dst] -= g (L2 f32 atomics).
// ---------------------------------------------------------------------------
__global__ void __launch_bounds__(256)
k_edge_force(const float* __restrict__ r, const float* __restrict__ radial,
             const int* __restrict__ src, const int* __restrict__ dst,
             const float* __restrict__ Fprime, float* __restrict__ forces) {
  __shared__ __align__(16) float s_coef[RAD_FLOATS];
  stage_radial_lds(s_coef, radial);

  const int stride = gridDim.x * blockDim.x;
  for (int e = blockIdx.x * blockDim.x + threadIdx.x; e < N_EDGES_C; e += stride) {
    if (e + stride < N_EDGES_C) __builtin_prefetch(r + 3 * (e + stride), 0, 0);
    const float x = r[3 * e + 0], y = r[3 * e + 1], z = r[3 * e + 2];
    const float bl = sqrtf(x * x + y * y + z * z);
    int idx = (int)(bl / DR_F);
    idx = idx < 0 ? 0 : (idx > N_INT - 1 ? N_INT - 1 : idx);
    const float dxv = bl - (float)idx * DR_F;
    const float4* c4 = (const float4*)(s_coef + idx * 8);
    const float4 A = c4[0];
    const float4 B = c4[1];
    const float rphi  = A.y + dxv * (A.w + dxv * (B.y + dxv * B.w));
    const float drho  = A.z + dxv * (2.0f * B.x + 3.0f * B.z * dxv);
    const float drphi = A.w + dxv * (2.0f * B.y + 3.0f * B.w * dxv);
    const int de = dst[e];
    const int se = src[e];
    const float fp = Fprime[de];
    const float inv_bl   = 1.0f / bl;
    const float half_inv = 0.5f * inv_bl;
    const float dphi = drphi * half_inv - rphi * half_inv * inv_bl;
    const float scale = (fp * drho + dphi) * inv_bl;
    const float gx = scale * x, gy = scale * y, gz = scale * z;
    atomicAdd(forces + 3 * se + 0, gx);
    atomicAdd(forces + 3 * se + 1, gy);
    atomicAdd(forces + 3 * se + 2, gz);
    atomicAdd(forces + 3 * de + 0, -gx);
    atomicAdd(forces + 3 * de + 1, -gy);
    atomicAdd(forces + 3 * de + 2, -gz);
  }
}

// ---------------------------------------------------------------------------
extern "C" void kernel_launch(void* const* d_in, const int* in_sizes, int n_in,
                              void* d_out, int out_size, void* d_ws, size_t ws_size,
                              hipStream_t stream) {
  const float* r      = (const float*)d_in[0];  // (3.2M, 3)
  const float* radial = (const float*)d_in[1];  // (2000, 4, 2)
  const float* embed  = (const float*)d_in[2];  // (2000, 4, 1)
  const int*   src    = (const int*)d_in[3];
  const int*   dst    = (const int*)d_in[4];

  float* out     = (float*)d_out;            // out[0] = energy, out[1..] = forces (50000,3)
  float* density = (float*)d_ws;             // 50000 f32
  float* Fprime  = density + N_NODES_C;      // 50000 f32

  const int nzero = 1 + 3 * N_NODES_C;       // covers d_out; density covered too (50000 < nzero)
  k_zero<<<(nzero + 255) / 256, 256, 0, stream>>>(out, density);
  k_edge_density<<<1024, 256, 0, stream>>>(r, radial, dst, density, out);
  k_embed<<<(N_NODES_C + 255) / 256, 256, 0, stream>>>(density, (const float4*)embed, Fprime, out);
  k_edge_force<<<1024, 256, 0, stream>>>(r, radial, src, dst, Fprime, out + 1);
}